// FixedPointLayer_52733608461007
// MI455X (gfx1250) — compile-verified
//
#include <hip/hip_runtime.h>
#include <cstdint>

// FixedPointLayer: x <- relu(GAMMA * A@x + b), H=4 heads, N=4096, fp32.
// Bandwidth-bound (A = 256 MB/application); 51 fixed applications ==
// reference while-loop semantics to < 0.5*tol. GEMV is expressed as
// V_WMMA_F32_16X16X4_F32 with x broadcast across the B-operand columns,
// x staged into LDS via gfx1250 async global->LDS loads.
//
// Round-1 fix: k-loop bounds made wave-uniform (pointers pre-offset by the
// split-K segment) so the compiler emits a scalar-counter loop instead of a
// divergent loop with per-iteration EXEC-mask bookkeeping.

#define NN 4096
#define NHALF (NN / 2)
#define HH 4
#define GAMMA 0.5f
#define NITER 51                       // MAX_ITER + 1 applications of f
#define ROWS_PER_BLOCK 64              // 4 row-tiles of 16, K split x2 -> 8 waves
#define BLOCKS_PER_HEAD (NN / ROWS_PER_BLOCK)   // 64
#define THREADS 256

typedef float v2f __attribute__((ext_vector_type(2)));
typedef float v8f __attribute__((ext_vector_type(8)));

__global__ void fp_zero_kernel(float* __restrict__ p, int n) {
  int i = blockIdx.x * blockDim.x + threadIdx.x;
  if (i < n) p[i] = 0.0f;
}

__global__ __launch_bounds__(THREADS)
void fp_iter_kernel(const float* __restrict__ A, const float* __restrict__ B,
                    const float* __restrict__ X, float* __restrict__ Y) {
  __shared__ __align__(16) float xs[NN];     // 16 KB: this head's x vector
  __shared__ float red[8 * 16];              // split-K partials, 8 waves x 16 rows

  const int h   = blockIdx.x / BLOCKS_PER_HEAD;
  const int blk = blockIdx.x % BLOCKS_PER_HEAD;
  const float* Ah = A + (size_t)h * NN * NN;
  const float* bh = B + (size_t)h * NN;
  const float* xh = X + (size_t)h * NN;
  float*       yh = Y + (size_t)h * NN;

  // ---- stage x into LDS with gfx1250 async tensor path ----
  // Generic pointers to __shared__ carry the LDS byte address in their low
  // 32 bits (flat->LDS aperture mapping), which is what VDST wants.
  {
    unsigned lbase = (unsigned)(unsigned long long)(const void*)xs;
    for (int off = (int)threadIdx.x * 16; off < NN * 4; off += THREADS * 16) {
      unsigned      laddr = lbase + (unsigned)off;
      unsigned long long gaddr =
          (unsigned long long)(uintptr_t)((const char*)xh + off);
      asm volatile("global_load_async_to_lds_b128 %0, %1, off"
                   :: "v"(laddr), "v"(gaddr) : "memory");
    }
    asm volatile("s_wait_asynccnt 0" ::: "memory");
  }
  __syncthreads();

  // ---- GEMV via WMMA f32 16x16x4 ----
  const int lane = threadIdx.x & 31;
  const int wave = threadIdx.x >> 5;
  const int tile = wave & 3;        // which 16-row tile within the 64-row block
  const int kseg = wave >> 2;       // split-K half: 0 or 1
  const int m    = lane & 15;       // row within tile
  const int half = lane >> 4;       // k sub-pair: lanes 0-15 -> k0,k1 ; 16-31 -> k2,k3

  const int row0   = blk * ROWS_PER_BLOCK + tile * 16;
  const int k_base = kseg * NHALF;  // fold split-K offset into the pointers so
                                    // the loop trip count is a literal constant
  const float* arow = Ah + (size_t)(row0 + m) * NN + k_base + (half << 1);
  const float* xrow = xs + k_base + (half << 1);

  v8f acc = {};
#pragma unroll 8
  for (int k = 0; k < NHALF; k += 4) {
    v2f a  = *(const v2f*)(arow + k);   // A[row, k..k+1 | k+2..k+3], global b64
    v2f xv = *(const v2f*)(xrow + k);   // x[k..k+1 | k+2..k+3] broadcast, LDS b64
    // D(16x16) = A_tile(16x4) x Xbcast(4x16) + C ; every column identical.
    acc = __builtin_amdgcn_wmma_f32_16x16x4_f32(
        /*neg_a=*/false, a, /*neg_b=*/false, xv,
        /*c_mod=*/(short)0, acc, /*reuse_a=*/false, /*reuse_b=*/false);
  }

  // Column 0 lives in lanes 0 (rows tile*16+0..7) and 16 (rows tile*16+8..15):
  // acc[d] = partial dot for local row tile*16 + half*8 + d.
  if (m == 0) {
#pragma unroll
    for (int d = 0; d < 8; ++d)
      red[wave * 16 + half * 8 + d] = acc[d];
  }
  __syncthreads();

  // Combine the two K halves, apply gamma, bias, relu; one thread per row.
  if (threadIdx.x < ROWS_PER_BLOCK) {
    int r    = (int)threadIdx.x;          // local row 0..63
    int t    = r >> 4;
    int rr   = r & 15;
    float s  = red[t * 16 + rr] + red[(t + 4) * 16 + rr];
    int row  = blk * ROWS_PER_BLOCK + r;
    yh[row]  = fmaxf(GAMMA * s + bh[row], 0.0f);
  }
}

extern "C" void kernel_launch(void* const* d_in, const int* in_sizes, int n_in,
                              void* d_out, int out_size, void* d_ws, size_t ws_size,
                              hipStream_t stream) {
  const float* A = (const float*)d_in[0];   // [H, N, N] fp32
  const float* B = (const float*)d_in[1];   // [H, N, 1] fp32
  float* buf0 = (float*)d_ws;               // ping-pong x buffers: 2 * H*N floats
  float* buf1 = buf0 + (size_t)HH * NN;

  fp_zero_kernel<<<dim3((HH * NN + THREADS - 1) / THREADS), dim3(THREADS), 0,
                   stream>>>(buf0, HH * NN);

  for (int it = 0; it < NITER; ++it) {
    const float* src = (it & 1) ? buf1 : buf0;
    float* dst = (it == NITER - 1) ? (float*)d_out
                                   : ((it & 1) ? buf0 : buf1);
    fp_iter_kernel<<<dim3(HH * BLOCKS_PER_HEAD), dim3(THREADS), 0, stream>>>(
        A, B, src, dst);
  }
}